// Augmentator_62088047231203
// MI455X (gfx1250) — compile-verified
//
#include <hip/hip_runtime.h>
#include <stdint.h>

namespace {
constexpr int S_LEN = 8192;   // samples per row (reference S)
constexpr int NTHR  = 256;    // 8 wave32 per workgroup
// max crop_len = floor(0.9*8192) = 7372; +1 for interp neighbor; +3 align slack
constexpr int LDSN  = 7376;   // 29,504 bytes of LDS per workgroup
constexpr int VPT   = 4;      // outputs per thread per iteration (float4 stores)
constexpr int NITER = S_LEN / (NTHR * VPT);   // 8
}

__global__ __launch_bounds__(NTHR) void augmentator_kernel(
    const float* __restrict__ x,
    const float* __restrict__ crop_u,
    const float* __restrict__ start_u,
    const int*   __restrict__ flip,
    float* __restrict__ out)
{
    __shared__ __align__(16) float tile[LDSN];

    const int b   = blockIdx.x;
    const int tid = threadIdx.x;

    // Per-row parameters (uniform across the workgroup -> scalar-float unit).
    const float ratio    = __fadd_rn(0.6f, __fmul_rn(0.3f, crop_u[b]));
    const int   crop_len = (int)floorf(__fmul_rn((float)S_LEN, ratio));
    const int   start    = (int)floorf(__fmul_rn(start_u[b], (float)(S_LEN - crop_len)));
    const float scale    = (float)(crop_len - 1) / (float)(S_LEN - 1);
    const int   do_flip  = flip[b];

    // Stage x[b, start4 .. start4+n_need) into LDS via async global->LDS DMA.
    const int start4 = start & ~3;                       // 16-byte aligned element base
    const int n_need = crop_len + 1 + (start - start4);  // elements needed in LDS
    const int nv     = (n_need + 3) >> 2;                // float4 vectors to fetch
    int nv_full      = (S_LEN - start4) >> 2;            // vectors fully inside the row
    if (nv_full > nv) nv_full = nv;

    const float* rowp = x + (size_t)b * (size_t)S_LEN;
    // Low 32 bits of the flat address of LDS == LDS byte offset (aperture mapping).
    const uint32_t lds_base = (uint32_t)(uintptr_t)(&tile[0]);

    // Bulk: 16 B per lane per issue, coalesced streaming read of the crop window.
    for (int v = tid; v < nv_full; v += NTHR) {
        uint32_t ldsa = lds_base + ((uint32_t)v << 4);
        uint32_t goff = (uint32_t)((start4 + (v << 2)) << 2);  // byte offset in row
        asm volatile("global_load_async_to_lds_b128 %0, %1, %2"
                     :: "v"(ldsa), "v"(goff), "s"(rowp) : "memory");
    }
    // Tail (<= ~8 elements): per-element, clamped to the row end.
    for (int e = (nv_full << 2) + tid; e < (nv << 2); e += NTHR) {
        int ge = start4 + e;
        if (ge > S_LEN - 1) ge = S_LEN - 1;   // clamped slots are never interpolated
        uint32_t ldsa = lds_base + ((uint32_t)e << 2);
        uint32_t goff = (uint32_t)(ge << 2);
        asm volatile("global_load_async_to_lds_b32 %0, %1, %2"
                     :: "v"(ldsa), "v"(goff), "s"(rowp) : "memory");
    }

    // Drain this wave's async copies, then sync the workgroup before LDS reads.
    asm volatile("s_wait_asynccnt 0" ::: "memory");
    __syncthreads();

    // Interpolate out of LDS; 4 consecutive outputs per thread -> b128 stores.
    // Invariants (no clamps needed): pos >= start => li >= 0;
    // i0 <= start+crop_len-1 <= S-2 => li <= crop_len+2 <= LDSN-2, i1 = i0+1 <= S-1.
    const float startf = (float)start;
    float* orow = out + (size_t)b * (size_t)S_LEN;

#define INTERP_ONE(jj, dst)                                                   \
    {                                                                         \
        float pos = __fadd_rn(startf, __fmul_rn((float)(jj), scale));         \
        int i0 = (int)floorf(pos);                                            \
        float f = __fsub_rn(pos, (float)i0);                                  \
        int li = i0 - start4;                                                 \
        if (li > LDSN - 2) li = LDSN - 2;  /* safety fence only */            \
        float v0 = tile[li];                                                  \
        float v1 = tile[li + 1];                                              \
        (dst) = __fadd_rn(__fmul_rn(v0, __fsub_rn(1.0f, f)),                  \
                          __fmul_rn(v1, f));                                  \
    }

    if (!do_flip) {
#pragma unroll
        for (int k = 0; k < NITER; ++k) {
            const int j0 = ((k * NTHR + tid) << 2);   // j0 % 4 == 0
            float r0, r1, r2, r3;
            INTERP_ONE(j0 + 0, r0)
            INTERP_ONE(j0 + 1, r1)
            INTERP_ONE(j0 + 2, r2)
            INTERP_ONE(j0 + 3, r3)
            *reinterpret_cast<float4*>(orow + j0) = make_float4(r0, r1, r2, r3);
        }
    } else {
#pragma unroll
        for (int k = 0; k < NITER; ++k) {
            const int j0 = ((k * NTHR + tid) << 2);
            float r0, r1, r2, r3;
            INTERP_ONE(j0 + 0, r0)
            INTERP_ONE(j0 + 1, r1)
            INTERP_ONE(j0 + 2, r2)
            INTERP_ONE(j0 + 3, r3)
            // outputs land at S-1-j0 .. S-4-j0 => reversed float4 at S-4-j0 (16B aligned)
            *reinterpret_cast<float4*>(orow + (S_LEN - VPT - j0)) =
                make_float4(r3, r2, r1, r0);
        }
    }
#undef INTERP_ONE
}

extern "C" void kernel_launch(void* const* d_in, const int* in_sizes, int n_in,
                              void* d_out, int out_size, void* d_ws, size_t ws_size,
                              hipStream_t stream) {
    const float* x       = (const float*)d_in[0];
    const float* crop_u  = (const float*)d_in[1];
    const float* start_u = (const float*)d_in[2];
    const int*   flip    = (const int*)d_in[3];
    float* out = (float*)d_out;
    const int B = in_sizes[1];   // 4096 rows (crop_u length)
    augmentator_kernel<<<dim3(B), dim3(NTHR), 0, stream>>>(x, crop_u, start_u, flip, out);
    (void)n_in; (void)out_size; (void)d_ws; (void)ws_size;
}